// BatchHardTripletLoss_28621662060785
// MI455X (gfx1250) — compile-verified
//
#include <hip/hip_runtime.h>
#include <hip/hip_bf16.h>

typedef float v2f __attribute__((ext_vector_type(2)));
typedef float v8f __attribute__((ext_vector_type(8)));

#define D 128
#define C 10

// ---------------- K1: row squared norms, one wave per row ----------------
__global__ void k1_x2(const float* __restrict__ E, float* __restrict__ x2, int N) {
    int wave = threadIdx.x >> 5, lane = threadIdx.x & 31;
    int row = blockIdx.x * 8 + wave;
    if (row >= N) return;
    float4 v = reinterpret_cast<const float4*>(E + (long)row * D)[lane];
    float s = v.x * v.x + v.y * v.y + v.z * v.z + v.w * v.w;
    #pragma unroll
    for (int m = 16; m >= 1; m >>= 1) s += __shfl_xor(s, m, 32);
    if (lane == 0) x2[row] = s;
}

// ------- K2: per-class count, sum of norms, centroid sum (deterministic) -------
__global__ void k2_classstats(const float* __restrict__ E, const int* __restrict__ lab,
                              const float* __restrict__ x2, float* __restrict__ M,
                              float* __restrict__ X2c, int* __restrict__ cnt, int N) {
    int c = blockIdx.x;      // class
    int d = threadIdx.x;     // dimension, 128 threads
    float msum = 0.0f, xp = 0.0f;
    int cp = 0;
    for (int r = 0; r < N; ++r) {
        int l = lab[r];
        if (l == c) {
            msum += E[(long)r * D + d];
            if ((r & (D - 1)) == d) { cp++; xp += x2[r]; }
        }
    }
    M[c * D + d] = msum;
    __shared__ float sx[D];
    __shared__ int sc[D];
    sx[d] = xp; sc[d] = cp;
    __syncthreads();
    #pragma unroll
    for (int s = D / 2; s > 0; s >>= 1) {
        if (d < s) { sx[d] += sx[d + s]; sc[d] += sc[d + s]; }
        __syncthreads();
    }
    if (d == 0) { X2c[c] = sx[0]; cnt[c] = sc[0]; }
}

// ------- K3: stable counting sort by (class, index); one wave per class -------
__global__ void k3_sort(const int* __restrict__ lab, const int* __restrict__ cnt,
                        int* __restrict__ order, int* __restrict__ lsorted,
                        int* __restrict__ offv, int N) {
    int c = blockIdx.x;
    int lane = threadIdx.x;   // 32 threads = 1 wave
    int off = 0;
    for (int cc = 0; cc < c; ++cc) off += cnt[cc];
    if (lane == 0) offv[c] = off;
    int base = 0;
    for (int r0 = 0; r0 < N; r0 += 32) {
        int r = r0 + lane;
        bool in = (r < N) && (lab[r] == c);
        unsigned mm = (unsigned)__ballot(in);
        int pos = __popc(mm & ((1u << lane) - 1u));
        if (in) {
            order[off + base + pos] = r;
            lsorted[off + base + pos] = c;
        }
        base += __popc(mm);
    }
}

// ------- K4: hardest-positive via fp32 WMMA over intra-class tiles -------
// Rows/cols in class-sorted order. WG = 16-row band, 8 waves stripe 16-col tiles.
// Dual accumulators break the WMMA D->C dependency chain (latency vs issue rate).
__global__ void __launch_bounds__(256) k4_dpos(const float* __restrict__ E,
                                               const float* __restrict__ x2g,
                                               const int* __restrict__ order,
                                               const int* __restrict__ lsorted,
                                               const int* __restrict__ cnt,
                                               const int* __restrict__ offv,
                                               float* __restrict__ dpos, int N) {
    __shared__ float wgmax[8][16];
    int wave = threadIdx.x >> 5, lane = threadIdx.x & 31;
    int half = lane >> 4, l16 = lane & 15;
    int r0 = blockIdx.x * 16;

    // A fragments: 16x4 f32 tile layout — lanes 0-15 rows M=0..15 hold K={k0,k0+1},
    // lanes 16-31 hold K={k0+2,k0+3}. Keep all 32 K-step fragments in registers.
    int oa = order[r0 + l16];
    const float* pa = E + (long)oa * D + 2 * half;
    v2f af[32];
    #pragma unroll
    for (int t = 0; t < 32; ++t) af[t] = *reinterpret_cast<const v2f*>(pa + 4 * t);

    // Per-lane metadata for its 8 output rows (m = rr + 8*half per C layout)
    float x2r[8];
    int labr[8];
    #pragma unroll
    for (int rr = 0; rr < 8; ++rr) {
        int rm = r0 + rr + 8 * half;
        x2r[rr] = x2g[order[rm]];
        labr[rr] = lsorted[rm];
    }

    // Column range: only the classes spanned by this row band (sorted => contiguous)
    int labFirst = lsorted[r0];
    int labLast = lsorted[r0 + 15];
    int jStart = offv[labFirst] & ~15;
    int jEnd = offv[labLast] + cnt[labLast];
    jEnd = (jEnd + 15) & ~15;       // N % 16 == 0 so this never exceeds N
    int nTiles = (jEnd - jStart) >> 4;

    float maxacc[8];
    #pragma unroll
    for (int rr = 0; rr < 8; ++rr) maxacc[rr] = -__builtin_inff();

    for (int idx = wave; idx < nTiles; idx += 8) {
        int j0 = jStart + (idx << 4);
        int crj = j0 + l16;
        int oc = order[crj];
        int lc = lsorted[crj];
        float xc = x2g[oc];
        const float* pb = E + (long)oc * D + 2 * half;
        v2f bf[32];
        #pragma unroll
        for (int t = 0; t < 32; ++t) bf[t] = *reinterpret_cast<const v2f*>(pb + 4 * t);

        v8f acc0 = {0.f, 0.f, 0.f, 0.f, 0.f, 0.f, 0.f, 0.f};
        v8f acc1 = {0.f, 0.f, 0.f, 0.f, 0.f, 0.f, 0.f, 0.f};
        #pragma unroll
        for (int t = 0; t < 32; t += 2) {
            acc0 = __builtin_amdgcn_wmma_f32_16x16x4_f32(false, af[t], false, bf[t],
                                                         (short)0, acc0, false, false);
            acc1 = __builtin_amdgcn_wmma_f32_16x16x4_f32(false, af[t + 1], false, bf[t + 1],
                                                         (short)0, acc1, false, false);
        }

        #pragma unroll
        for (int rr = 0; rr < 8; ++rr) {
            if (labr[rr] == lc) {   // same-class mask (also masks cross-class boundary cols)
                float g = acc0[rr] + acc1[rr];
                float dd = fmaxf(x2r[rr] + xc - 2.0f * g, 0.0f);
                maxacc[rr] = fmaxf(maxacc[rr], dd);
            }
        }
    }

    // Reduce the 16 column-lanes of each half (masks 1..8 keep halves separate)
    #pragma unroll
    for (int m = 8; m >= 1; m >>= 1) {
        #pragma unroll
        for (int rr = 0; rr < 8; ++rr)
            maxacc[rr] = fmaxf(maxacc[rr], __shfl_xor(maxacc[rr], m, 32));
    }
    if (l16 == 0) {
        #pragma unroll
        for (int rr = 0; rr < 8; ++rr) wgmax[wave][rr + 8 * half] = maxacc[rr];
    }
    __syncthreads();
    if (threadIdx.x < 16) {
        int m = threadIdx.x;
        float v = -__builtin_inff();
        #pragma unroll
        for (int w = 0; w < 8; ++w) v = fmaxf(v, wgmax[w][m]);
        dpos[order[r0 + m]] = v;   // every global row owned by exactly one WG
    }
}

// ------- K5: per-anchor (wave each): S[c] via centroids, argmin, j*, d_neg, hinge -------
__global__ void __launch_bounds__(256) k5_anchor(const float* __restrict__ E,
                                                 const int* __restrict__ lab,
                                                 const float* __restrict__ x2,
                                                 const float* __restrict__ Mg,
                                                 const float* __restrict__ X2c,
                                                 const int* __restrict__ cnt,
                                                 const int* __restrict__ offv,
                                                 const int* __restrict__ order,
                                                 const float* __restrict__ dpos,
                                                 float* __restrict__ loss, int N) {
    __shared__ float4 M4[C * (D / 4)];
    __shared__ float sX2c[C];
    __shared__ int scnt[C], soff[C];
    for (int t = threadIdx.x; t < C * (D / 4); t += 256)
        M4[t] = reinterpret_cast<const float4*>(Mg)[t];
    if (threadIdx.x < C) {
        sX2c[threadIdx.x] = X2c[threadIdx.x];
        scnt[threadIdx.x] = cnt[threadIdx.x];
        soff[threadIdx.x] = offv[threadIdx.x];
    }
    __syncthreads();

    int wave = threadIdx.x >> 5, lane = threadIdx.x & 31;
    int i = blockIdx.x * 8 + wave;
    if (i >= N) return;

    float4 e = reinterpret_cast<const float4*>(E + (long)i * D)[lane];
    float dotc[C];
    #pragma unroll
    for (int c = 0; c < C; ++c) {
        float4 m = M4[c * (D / 4) + lane];
        float p = e.x * m.x + e.y * m.y + e.z * m.z + e.w * m.w;
        #pragma unroll
        for (int s = 16; s >= 1; s >>= 1) p += __shfl_xor(p, s, 32);
        dotc[c] = p;   // uniform across the wave after reduction
    }

    float xi = x2[i];
    int L = lab[i];
    float S[C], T = 0.0f;
    #pragma unroll
    for (int c = 0; c < C; ++c) {
        S[c] = (float)scnt[c] * xi + sX2c[c] - 2.0f * dotc[c];
        T += S[c];
    }
    // first-occurrence argmin over neg_dist[c] = T - S[c]
    float best = __builtin_inff();
    int kstar = 0;
    #pragma unroll
    for (int c = 0; c < C; ++c) {
        float nd = T - S[c];
        if (nd < best) { best = nd; kstar = c; }
    }
    // (k*+1)-th negative in (class, index) order -> class c*, rank r
    int k = kstar, cstar = -1, rrank = 0;
    #pragma unroll
    for (int c = 0; c < C; ++c) {
        if (c == L) continue;
        if (cstar < 0) {
            if (k < scnt[c]) { cstar = c; rrank = k; }
            else k -= scnt[c];
        }
    }
    int jstar = (cstar >= 0) ? order[soff[cstar] + rrank] : 0;

    float4 g = reinterpret_cast<const float4*>(E + (long)jstar * D)[lane];
    float pn = e.x * g.x + e.y * g.y + e.z * g.z + e.w * g.w;
    #pragma unroll
    for (int s = 16; s >= 1; s >>= 1) pn += __shfl_xor(pn, s, 32);

    if (lane == 0) {
        float dn = fmaxf(xi + x2[jstar] - 2.0f * pn, 0.0f);
        loss[i] = fmaxf(dpos[i] - dn + 1.0f, 0.0f);   // MARGIN = 1.0
    }
}

// ---------------- K6: deterministic mean reduction ----------------
__global__ void k6_reduce(const float* __restrict__ loss, float* __restrict__ out, int N) {
    __shared__ float s[256];
    float p = 0.0f;
    for (int r = threadIdx.x; r < N; r += 256) p += loss[r];
    s[threadIdx.x] = p;
    __syncthreads();
    for (int t = 128; t > 0; t >>= 1) {
        if (threadIdx.x < t) s[threadIdx.x] += s[threadIdx.x + t];
        __syncthreads();
    }
    if (threadIdx.x == 0) out[0] = s[0] / (float)N;
}

extern "C" void kernel_launch(void* const* d_in, const int* in_sizes, int n_in,
                              void* d_out, int out_size, void* d_ws, size_t ws_size,
                              hipStream_t stream) {
    const float* E = (const float*)d_in[0];   // [N, 128] fp32
    const int* lab = (const int*)d_in[1];     // [N] int32
    int N = in_sizes[1];                      // 8192

    // Workspace layout (all region starts 16B-aligned; N % 16 == 0)
    float* ws = (float*)d_ws;
    float* x2 = ws;                    // N
    float* dpos = ws + N;              // N
    float* loss = ws + 2 * N;          // N
    float* M = ws + 3 * N;             // C*D = 1280
    float* X2c = M + C * D;            // 16 slots
    int* order = (int*)(X2c + 16);     // N
    int* lsorted = order + N;          // N
    int* cnt = lsorted + N;            // 16 slots
    int* offv = cnt + 16;              // 16 slots

    k1_x2<<<N / 8, 256, 0, stream>>>(E, x2, N);
    k2_classstats<<<C, D, 0, stream>>>(E, lab, x2, M, X2c, cnt, N);
    k3_sort<<<C, 32, 0, stream>>>(lab, cnt, order, lsorted, offv, N);
    k4_dpos<<<N / 16, 256, 0, stream>>>(E, x2, order, lsorted, cnt, offv, dpos, N);
    k5_anchor<<<N / 8, 256, 0, stream>>>(E, lab, x2, M, X2c, cnt, offv, order, dpos, loss, N);
    k6_reduce<<<1, 256, 0, stream>>>(loss, (float*)d_out, N);
}